// LinearAttention_66949950210409
// MI455X (gfx1250) — compile-verified
//
#include <hip/hip_runtime.h>
#include <hip/hip_bf16.h>

// Linear attention for MI455X (gfx1250), fp32 end-to-end using
// V_WMMA_F32_16X16X4_F32. Memory-bound (~1.3GB traffic vs ~283 GFLOP), so fp32
// matrix cores are the right precision: exact vs reference, far below ceiling.
//
// WMMA wave32 layouts (ISA 7.12.2):
//   A (16x4, MxK):  lane = 16*kh + m (kh=lane>>4, m=lane&15); vgpr v holds K = 2*kh+v
//   B (4x16, KxN):  lane = 16*kh + n;  vgpr v holds K = 2*kh+v
//   C/D (16x16):    lane = 16*mh + n;  vgpr r holds M = r + 8*mh
//
// Data movement: contiguous / row-pitch tiles are staged with
// GLOBAL_LOAD_ASYNC_TO_LDS_B128 (ASYNCcnt) — bypasses VGPRs entirely;
// transposed stages (W tiles, k/v in k2) keep the VGPR+ds_store path.

typedef float v2f __attribute__((ext_vector_type(2)));
typedef float v8f __attribute__((ext_vector_type(8)));

#define B_  4
#define L_  8192
#define D_  1024
#define H_  16
#define DH  64
#define EPSF 1e-6f

#if __has_builtin(__builtin_amdgcn_global_load_async_to_lds_b128)
#define HAS_ASYNC_LDS 1
typedef int vi4 __attribute__((vector_size(16)));
typedef __attribute__((address_space(1))) vi4 as1_vi4;   // global
typedef __attribute__((address_space(3))) vi4 as3_vi4;   // LDS
#else
#define HAS_ASYNC_LDS 0
#endif

__device__ __forceinline__ void async_copy16(const float* g, float* l) {
#if HAS_ASYNC_LDS
  __builtin_amdgcn_global_load_async_to_lds_b128((as1_vi4*)g, (as3_vi4*)l, 0, 0);
#else
  *(float4*)l = *(const float4*)g;
#endif
}

__device__ __forceinline__ void async_wait() {
#if HAS_ASYNC_LDS
#if __has_builtin(__builtin_amdgcn_s_wait_asynccnt)
  __builtin_amdgcn_s_wait_asynccnt(0);
#else
  asm volatile("s_wait_asynccnt 0" ::: "memory");
#endif
#endif
}

__device__ __forceinline__ v8f wmma4(v2f a, v2f b, v8f c) {
  return __builtin_amdgcn_wmma_f32_16x16x4_f32(
      /*neg_a=*/false, a, /*neg_b=*/false, b,
      /*c_mod=*/(short)0, c, /*reuse_a=*/false, /*reuse_b=*/false);
}

__device__ __forceinline__ v8f v8f_zero() {
  v8f z;
  for (int e = 0; e < 8; ++e) z[e] = 0.0f;
  return z;
}

__device__ __forceinline__ float phi_f(float x) {
  // elu(x)+1 == x+1 (x>0) else exp(x)
  return x > 0.0f ? x + 1.0f : __expf(x);
}

// ---------------------------------------------------------------------------
// Kernel 1: qkv = x @ Wqkv  (32768x1024 @ 1024x3072)
// Block tile 128x256, 8 waves of 64x64. Epilogue: phi on q,k; scatter to
// per-head layout ws[sec][b][h][l][d], sec in {q,k,v}.
// ---------------------------------------------------------------------------
__global__ __launch_bounds__(256)
void k1_qkv(const float* __restrict__ X, const float* __restrict__ Wqkv,
            float* __restrict__ qkv_ws) {
  __shared__ __attribute__((aligned(16))) float lds_x[128 * 16];   // [row][k]
  __shared__ __attribute__((aligned(16))) float lds_w[256 * 16];   // [n][k]

  const int tid  = threadIdx.x;
  const int wave = tid >> 5, lane = tid & 31;
  const int half = lane >> 4, ln = lane & 15;
  const int wm = (wave & 1) * 64, wn = (wave >> 1) * 64;
  const int rowBase = blockIdx.x * 128;
  const int colBase = blockIdx.y * 256;

  v8f acc[4][4];
  for (int i = 0; i < 4; ++i)
    for (int j = 0; j < 4; ++j) acc[i][j] = v8f_zero();

  for (int kt = 0; kt < 1024; kt += 16) {
    // X tile 128x16: async global->LDS, per-lane b128 (2 per thread)
    for (int u = 0; u < 2; ++u) {
      int f = tid + u * 256;
      int r = f >> 2, c4 = (f & 3) << 2;
      async_copy16(&X[(size_t)(rowBase + r) * 1024 + kt + c4],
                   &lds_x[r * 16 + c4]);
    }
    // Prefetch next W tile into cache hierarchy
    if (kt + 16 < 1024) {
      int f = tid;
      int kr = f >> 6, c4 = (f & 63) << 2;
      __builtin_prefetch(&Wqkv[(size_t)(kt + 16 + kr) * 3072 + colBase + c4], 0, 1);
    }
    // W tile 16x256 staged transposed via VGPRs (1024 float4, 4 per thread)
    for (int u = 0; u < 4; ++u) {
      int f = tid + u * 256;
      int kr = f >> 6, c4 = (f & 63) << 2;
      float4 wv = *(const float4*)&Wqkv[(size_t)(kt + kr) * 3072 + colBase + c4];
      lds_w[(c4 + 0) * 16 + kr] = wv.x;
      lds_w[(c4 + 1) * 16 + kr] = wv.y;
      lds_w[(c4 + 2) * 16 + kr] = wv.z;
      lds_w[(c4 + 3) * 16 + kr] = wv.w;
    }
    async_wait();
    __syncthreads();
    for (int kk = 0; kk < 16; kk += 4) {
      v2f a[4], b[4];
      for (int i = 0; i < 4; ++i)
        a[i] = *(const v2f*)&lds_x[(wm + i * 16 + ln) * 16 + kk + 2 * half];
      for (int j = 0; j < 4; ++j)
        b[j] = *(const v2f*)&lds_w[(wn + j * 16 + ln) * 16 + kk + 2 * half];
      for (int i = 0; i < 4; ++i)
        for (int j = 0; j < 4; ++j)
          acc[i][j] = wmma4(a[i], b[j], acc[i][j]);
    }
    __syncthreads();
  }

  // Epilogue: phi on q,k; scatter into [sec][b][h][l][d]
  const size_t SEC = (size_t)B_ * H_ * L_ * DH;
  size_t jbase[4];
  int    jsec[4];
  for (int j = 0; j < 4; ++j) {
    int col = colBase + wn + j * 16 + ln;
    int sec = col >> 10, cd = col & 1023;
    int h = cd >> 6, d = cd & 63;
    jsec[j]  = sec;
    jbase[j] = (size_t)sec * SEC + (size_t)h * ((size_t)L_ * DH) + d;
  }
  for (int i = 0; i < 4; ++i) {
    for (int r = 0; r < 8; ++r) {
      int row = rowBase + wm + i * 16 + r + 8 * half;
      int bb = row >> 13, l = row & 8191;
      size_t rt = (size_t)bb * ((size_t)H_ * L_ * DH) + (size_t)l * DH;
      for (int j = 0; j < 4; ++j) {
        float v = acc[i][j][r];
        if (jsec[j] < 2) v = phi_f(v);
        qkv_ws[jbase[j] + rt] = v;
      }
    }
  }
}

// ---------------------------------------------------------------------------
// Kernel 2: per (b,h): kv[64][64] = sum_l (m*k)[l]^T (m*v)[l], ksum[64].
// 64 blocks; 8 waves each own two 16x16 kv tiles; full workgroup stages a
// 64-row L chunk transposed into LDS with mask applied.
// kv is written TRANSPOSED ([n][m]) so kernel 3 can stage it contiguously.
// ---------------------------------------------------------------------------
__global__ __launch_bounds__(256)
void k2_kv(const float* __restrict__ qkv_ws, const int* __restrict__ mask,
           float* __restrict__ kv_ws, float* __restrict__ ksum_ws) {
  __shared__ __attribute__((aligned(16))) float lds_kT[64 * 64];   // [d][l]
  __shared__ __attribute__((aligned(16))) float lds_vT[64 * 64];   // [d][l]

  const int tid  = threadIdx.x;
  const int wave = tid >> 5, lane = tid & 31;
  const int half = lane >> 4, ln = lane & 15;
  const int bh = blockIdx.x, b = bh >> 4;

  const size_t SEC = (size_t)B_ * H_ * L_ * DH;
  const float* kp = qkv_ws + SEC + (size_t)bh * L_ * DH;
  const float* vp = qkv_ws + 2 * SEC + (size_t)bh * L_ * DH;

  const int iM = wave >> 1;          // M tile (k's d)
  const int j0 = (wave & 1) * 2;     // first of two N tiles (v's d)

  v8f acc[2];
  acc[0] = v8f_zero();
  acc[1] = v8f_zero();
  float ksacc = 0.0f;

  for (int c = 0; c < 128; ++c) {
    int l0 = c * 64;
    for (int u = 0; u < 4; ++u) {
      int f = tid + u * 256;
      int ll = f >> 4, c4 = (f & 15) << 2;
      float m = (float)mask[b * L_ + l0 + ll];
      float4 kv4 = *(const float4*)&kp[(size_t)(l0 + ll) * DH + c4];
      lds_kT[(c4 + 0) * 64 + ll] = kv4.x * m;
      lds_kT[(c4 + 1) * 64 + ll] = kv4.y * m;
      lds_kT[(c4 + 2) * 64 + ll] = kv4.z * m;
      lds_kT[(c4 + 3) * 64 + ll] = kv4.w * m;
      float4 vv4 = *(const float4*)&vp[(size_t)(l0 + ll) * DH + c4];
      lds_vT[(c4 + 0) * 64 + ll] = vv4.x * m;
      lds_vT[(c4 + 1) * 64 + ll] = vv4.y * m;
      lds_vT[(c4 + 2) * 64 + ll] = vv4.z * m;
      lds_vT[(c4 + 3) * 64 + ll] = vv4.w * m;
    }
    __syncthreads();
    // ksum: waves 0..1 (tid<64) own one d each
    if (tid < 64) {
      for (int l4 = 0; l4 < 64; l4 += 4) {
        float4 t = *(const float4*)&lds_kT[tid * 64 + l4];
        ksacc += t.x + t.y + t.z + t.w;
      }
    }
    for (int kk = 0; kk < 64; kk += 4) {
      v2f a  = *(const v2f*)&lds_kT[(iM * 16 + ln) * 64 + kk + 2 * half];
      v2f b0 = *(const v2f*)&lds_vT[((j0 + 0) * 16 + ln) * 64 + kk + 2 * half];
      v2f b1 = *(const v2f*)&lds_vT[((j0 + 1) * 16 + ln) * 64 + kk + 2 * half];
      acc[0] = wmma4(a, b0, acc[0]);
      acc[1] = wmma4(a, b1, acc[1]);
    }
    __syncthreads();
  }

  // kv stored transposed: kv_ws[bh][n][m]
  for (int jj = 0; jj < 2; ++jj) {
    for (int r = 0; r < 8; ++r) {
      int m = iM * 16 + r + 8 * half;
      int n = (j0 + jj) * 16 + ln;
      kv_ws[(size_t)bh * 4096 + n * 64 + m] = acc[jj][r];
    }
  }
  if (tid < 64) ksum_ws[bh * 64 + tid] = ksacc;
}

// ---------------------------------------------------------------------------
// Kernel 3: per (b,h, 128-row l tile): y = (q @ kv) * z, z = 1/(q.ksum + eps).
// 4096 blocks; each of 8 waves owns 16 rows x 64 cols.
// q / kv^T / ksum tiles are contiguous in memory -> async global->LDS copies.
// Writes y directly in [B,L,D] layout for the final GEMM.
// ---------------------------------------------------------------------------
__global__ __launch_bounds__(256)
void k3_y(const float* __restrict__ qkv_ws, const float* __restrict__ kv_ws,
          const float* __restrict__ ksum_ws, float* __restrict__ y_ws) {
  __shared__ __attribute__((aligned(16))) float lds_q[128 * 64];    // [l][d]
  __shared__ __attribute__((aligned(16))) float lds_kvT[64 * 64];   // [n][d]
  __shared__ __attribute__((aligned(16))) float lds_ks[64];
  __shared__ __attribute__((aligned(16))) float lds_z[128];

  const int tid  = threadIdx.x;
  const int wave = tid >> 5, lane = tid & 31;
  const int half = lane >> 4, ln = lane & 15;
  const int bh = blockIdx.x >> 6;
  const int l0 = (blockIdx.x & 63) * 128;
  const int b = bh >> 4, h = bh & 15;

  const float* qp = qkv_ws + (size_t)bh * L_ * DH;

  // q tile: 8192 contiguous floats
  for (int u = 0; u < 8; ++u) {
    int f = (tid + u * 256) << 2;
    async_copy16(&qp[(size_t)l0 * DH + f], &lds_q[f]);
  }
  // kv^T tile: 4096 contiguous floats (already [n][m] in memory)
  for (int u = 0; u < 4; ++u) {
    int f = (tid + u * 256) << 2;
    async_copy16(&kv_ws[(size_t)bh * 4096 + f], &lds_kvT[f]);
  }
  // ksum: 64 contiguous floats
  if (tid < 16)
    async_copy16(&ksum_ws[bh * 64 + tid * 4], &lds_ks[tid * 4]);
  async_wait();
  __syncthreads();

  if (tid < 128) {
    float s = 0.0f;
    for (int d4 = 0; d4 < 64; d4 += 4) {
      float4 qv = *(const float4*)&lds_q[tid * 64 + d4];
      float4 kv = *(const float4*)&lds_ks[d4];
      s += qv.x * kv.x + qv.y * kv.y + qv.z * kv.z + qv.w * kv.w;
    }
    lds_z[tid] = 1.0f / (s + EPSF);
  }
  __syncthreads();

  v8f acc[4];
  for (int j = 0; j < 4; ++j) acc[j] = v8f_zero();
  for (int kk = 0; kk < 64; kk += 4) {
    v2f a = *(const v2f*)&lds_q[(wave * 16 + ln) * 64 + kk + 2 * half];
    for (int j = 0; j < 4; ++j) {
      v2f bb = *(const v2f*)&lds_kvT[(j * 16 + ln) * 64 + kk + 2 * half];
      acc[j] = wmma4(a, bb, acc[j]);
    }
  }

  for (int r = 0; r < 8; ++r) {
    int ml = wave * 16 + r + 8 * half;
    float z = lds_z[ml];
    int l = l0 + ml;
    size_t rowOff = ((size_t)b * L_ + l) * D_ + h * DH;
    for (int j = 0; j < 4; ++j)
      y_ws[rowOff + j * 16 + ln] = acc[j][r] * z;
  }
}

// ---------------------------------------------------------------------------
// Kernel 4: out = (y @ Wout) * mask  (32768x1024 @ 1024x1024)
// ---------------------------------------------------------------------------
__global__ __launch_bounds__(256)
void k4_out(const float* __restrict__ Y, const float* __restrict__ Wout,
            const int* __restrict__ mask, float* __restrict__ out) {
  __shared__ __attribute__((aligned(16))) float lds_y[128 * 16];   // [row][k]
  __shared__ __attribute__((aligned(16))) float lds_w[256 * 16];   // [n][k]

  const int tid  = threadIdx.x;
  const int wave = tid >> 5, lane = tid & 31;
  const int half = lane >> 4, ln = lane & 15;
  const int wm = (wave & 1) * 64, wn = (wave >> 1) * 64;
  const int rowBase = blockIdx.x * 128;
  const int colBase = blockIdx.y * 256;

  v8f acc[4][4];
  for (int i = 0; i < 4; ++i)
    for (int j = 0; j < 4; ++j) acc[i][j] = v8f_zero();

  for (int kt = 0; kt < 1024; kt += 16) {
    for (int u = 0; u < 2; ++u) {
      int f = tid + u * 256;
      int r = f >> 2, c4 = (f & 3) << 2;
      async_copy16(&Y[(size_t)(rowBase + r) * 1024 + kt + c4],
                   &lds_y[r * 16 + c4]);
    }
    if (kt + 16 < 1024) {
      int f = tid;
      int kr = f >> 6, c4 = (f & 63) << 2;
      __builtin_prefetch(&Wout[(size_t)(kt + 16 + kr) * 1024 + colBase + c4], 0, 1);
    }
    for (int u = 0; u < 4; ++u) {
      int f = tid + u * 256;
      int kr = f >> 6, c4 = (f & 63) << 2;
      float4 wv = *(const float4*)&Wout[(size_t)(kt + kr) * 1024 + colBase + c4];
      lds_w[(c4 + 0) * 16 + kr] = wv.x;
      lds_w[(c4 + 1) * 16 + kr] = wv.y;
      lds_w[(c4 + 2) * 16 + kr] = wv.z;
      lds_w[(c4 + 3) * 16 + kr] = wv.w;
    }
    async_wait();
    __syncthreads();
    for (int kk = 0; kk < 16; kk += 4) {
      v2f a[4], b[4];
      for (int i = 0; i < 4; ++i)
        a[i] = *(const v2f*)&lds_y[(wm + i * 16 + ln) * 16 + kk + 2 * half];
      for (int j = 0; j < 4; ++j)
        b[j] = *(const v2f*)&lds_w[(wn + j * 16 + ln) * 16 + kk + 2 * half];
      for (int i = 0; i < 4; ++i)
        for (int j = 0; j < 4; ++j)
          acc[i][j] = wmma4(a[i], b[j], acc[i][j]);
    }
    __syncthreads();
  }

  for (int i = 0; i < 4; ++i) {
    for (int r = 0; r < 8; ++r) {
      int row = rowBase + wm + i * 16 + r + 8 * half;
      float m = (float)mask[row];
      for (int j = 0; j < 4; ++j) {
        int col = colBase + wn + j * 16 + ln;
        out[(size_t)row * 1024 + col] = acc[i][j][r] * m;
      }
    }
  }
}

// ---------------------------------------------------------------------------
extern "C" void kernel_launch(void* const* d_in, const int* in_sizes, int n_in,
                              void* d_out, int out_size, void* d_ws, size_t ws_size,
                              hipStream_t stream) {
  (void)in_sizes; (void)n_in; (void)out_size; (void)ws_size;
  const float* x    = (const float*)d_in[0];
  const int*   mask = (const int*)d_in[1];
  const float* Wqkv = (const float*)d_in[2];
  const float* Wout = (const float*)d_in[3];
  float* out = (float*)d_out;
  float* ws  = (float*)d_ws;

  const size_t SEC = (size_t)B_ * H_ * L_ * DH;   // 33,554,432 floats
  float* qkv_ws  = ws;                            // 3*SEC  (q, k, v per-head)
  float* kv_ws   = ws + 3 * SEC;                  // 64*4096 (transposed [n][m])
  float* ksum_ws = kv_ws + (size_t)64 * 4096;     // 64*64
  float* y_ws    = ksum_ws + 64 * 64;             // 32768*1024

  k1_qkv<<<dim3(256, 12), 256, 0, stream>>>(x, Wqkv, qkv_ws);
  k2_kv<<<64, 256, 0, stream>>>(qkv_ws, mask, kv_ws, ksum_ws);
  k3_y<<<4096, 256, 0, stream>>>(qkv_ws, kv_ws, ksum_ws, y_ws);
  k4_out<<<dim3(256, 4), 256, 0, stream>>>(y_ws, Wout, mask, out);
}